// GATLayer_10118942949376
// MI455X (gfx1250) — compile-verified
//
#include <hip/hip_runtime.h>
#include <math.h>

#define N_NODES 6144
#define IN_F    256
#define OUT_F   64
#define HEADS   4

typedef __attribute__((ext_vector_type(2))) float v2f;
typedef __attribute__((ext_vector_type(8))) float v8f;

__device__ __forceinline__ v8f wmma_f32_16x16x4(v2f a, v2f b, v8f c) {
  // D = A(16x4) x B(4x16) + C(16x16), all f32, wave32
  return __builtin_amdgcn_wmma_f32_16x16x4_f32(false, a, false, b, (short)0, c, false, false);
}

// ---------------------------------------------------------------------------
// Kernel 1: h[head][n][o] = sum_i x[n][i] * W[head][i][o]
// One wave per 16-row stripe; 4 f32 accumulators cover all 64 output columns.
// ---------------------------------------------------------------------------
__global__ void gat_h_kernel(const float* __restrict__ x, const float* __restrict__ W,
                             float* __restrict__ h) {
  const int head = blockIdx.y;
  const int wave = threadIdx.x >> 5;
  const int lane = threadIdx.x & 31;
  const int half = lane >> 4;   // 0: K=k,k+1   1: K=k+2,k+3
  const int l    = lane & 15;   // M index for A, N index for B/C/D
  const int n0   = blockIdx.x * 64 + wave * 16;

  const float* Wh   = W + (size_t)head * IN_F * OUT_F;
  const float* xrow = x + (size_t)(n0 + l) * IN_F;

  v8f acc0 = {}, acc1 = {}, acc2 = {}, acc3 = {};
  for (int k = 0; k < IN_F; k += 4) {
    const int ka = k + half * 2;
    v2f a;
    a.x = xrow[ka];
    a.y = xrow[ka + 1];
    const float* wk0 = Wh + (size_t)ka * OUT_F + l;
    const float* wk1 = wk0 + OUT_F;
    v2f b0; b0.x = wk0[0];  b0.y = wk1[0];
    v2f b1; b1.x = wk0[16]; b1.y = wk1[16];
    v2f b2; b2.x = wk0[32]; b2.y = wk1[32];
    v2f b3; b3.x = wk0[48]; b3.y = wk1[48];
    acc0 = wmma_f32_16x16x4(a, b0, acc0);
    acc1 = wmma_f32_16x16x4(a, b1, acc1);
    acc2 = wmma_f32_16x16x4(a, b2, acc2);
    acc3 = wmma_f32_16x16x4(a, b3, acc3);
  }
  float* hb = h + (size_t)head * N_NODES * OUT_F;
  for (int r = 0; r < 8; ++r) {
    const int m = n0 + r + half * 8;   // D layout: VGPR r -> M=r (lanes 0-15) / r+8
    float* row = hb + (size_t)m * OUT_F + l;
    row[0]  = acc0[r];
    row[16] = acc1[r];
    row[32] = acc2[r];
    row[48] = acc3[r];
  }
}

// ---------------------------------------------------------------------------
// Kernel 2: per-(head,node) attention scalars + factored exponentials.
// ssrc = h.a_src ; pk[t] = {sdst, exp(sdst), exp(0.2*sdst), 0}
// ---------------------------------------------------------------------------
__global__ void gat_scores_kernel(const float* __restrict__ h, const float* __restrict__ a_src,
                                  const float* __restrict__ a_dst, float* __restrict__ ssrc,
                                  float4* __restrict__ pk) {
  const int t = blockIdx.x * blockDim.x + threadIdx.x;
  if (t >= HEADS * N_NODES) return;
  const int head = t / N_NODES;
  const int n    = t - head * N_NODES;
  const float4* hr = (const float4*)(h + ((size_t)head * N_NODES + n) * OUT_F);
  const float4* as = (const float4*)(a_src + head * OUT_F);
  const float4* ad = (const float4*)(a_dst + head * OUT_F);
  float s = 0.f, d = 0.f;
#pragma unroll
  for (int o = 0; o < OUT_F / 4; ++o) {
    const float4 hv = hr[o];
    const float4 sv = as[o];
    const float4 dv = ad[o];
    s += hv.x * sv.x + hv.y * sv.y + hv.z * sv.z + hv.w * sv.w;
    d += hv.x * dv.x + hv.y * dv.y + hv.z * dv.z + hv.w * dv.w;
  }
  ssrc[t] = s;
  float4 p;
  p.x = d;
  p.y = __expf(d);
  p.z = __expf(0.2f * d);
  p.w = 0.f;
  pk[t] = p;
}

// ---------------------------------------------------------------------------
// Kernel 3: masked-softmax row statistics. One block per row; adj streamed
// once as b128 loads; all 4 heads accumulated in the same pass.
// ---------------------------------------------------------------------------
__global__ void gat_rowstats_kernel(const int* __restrict__ adj, const float* __restrict__ ssrc,
                                    const float4* __restrict__ pk, float* __restrict__ C1,
                                    float* __restrict__ C5) {
  const int n   = blockIdx.x;
  const int tid = threadIdx.x;
  const int* arow = adj + (size_t)n * N_NODES;

  float th[HEADS], mmax[HEADS], s1[HEADS], s5[HEADS];
#pragma unroll
  for (int hh = 0; hh < HEADS; ++hh) {
    th[hh]   = -ssrc[hh * N_NODES + n];
    mmax[hh] = -3.4e38f;
    s1[hh]   = 0.f;
    s5[hh]   = 0.f;
  }
  // 256 threads x int4 -> 1024 adj entries per trip, 6144/1024 = 6 trips
  for (int m0 = tid * 4; m0 < N_NODES; m0 += 1024) {
    __builtin_prefetch(arow + m0 + 8192, 0, 0);   // global_prefetch_b8
    const int4 av = *(const int4*)(arow + m0);
    const int a4[4] = {av.x, av.y, av.z, av.w};
#pragma unroll
    for (int j = 0; j < 4; ++j) {
      if (a4[j] != 0) {
#pragma unroll
        for (int hh = 0; hh < HEADS; ++hh) {
          const float4 p = pk[hh * N_NODES + m0 + j];
          mmax[hh] = fmaxf(mmax[hh], p.x);
          if (p.x >= th[hh]) s1[hh] += p.y;
          else               s5[hh] += p.z;
        }
      }
    }
  }

  __shared__ float red[3][HEADS][256];
#pragma unroll
  for (int hh = 0; hh < HEADS; ++hh) {
    red[0][hh][tid] = mmax[hh];
    red[1][hh][tid] = s1[hh];
    red[2][hh][tid] = s5[hh];
  }
  __syncthreads();
  for (int s = 128; s > 0; s >>= 1) {
    if (tid < s) {
#pragma unroll
      for (int hh = 0; hh < HEADS; ++hh) {
        red[0][hh][tid] = fmaxf(red[0][hh][tid], red[0][hh][tid + s]);
        red[1][hh][tid] += red[1][hh][tid + s];
        red[2][hh][tid] += red[2][hh][tid + s];
      }
    }
    __syncthreads();
  }
  if (tid < HEADS) {
    const int hh = tid;
    const float mm = red[0][hh][0];
    float c1 = 0.f, c5 = 0.f;
    if (mm > -3.0e38f) {                 // row has neighbors
      const float ss   = -th[hh];        // ssrc
      const float t    = ss + mm;
      const float rmax = (t >= 0.f) ? t : 0.2f * t;  // lrelu monotone -> row max
      const float e1   = __expf(ss - rmax);
      const float e5   = __expf(0.2f * ss - rmax);
      const float rsum = e1 * red[1][hh][0] + e5 * red[2][hh][0];
      const float inv  = (rsum > 0.f) ? (1.f / rsum) : 0.f;
      c1 = e1 * inv;
      c5 = e5 * inv;
    }
    C1[hh * N_NODES + n] = c1;
    C5[hh * N_NODES + n] = c5;
  }
}

// ---------------------------------------------------------------------------
// Kernel 4 (fused): materialize attn A-fragments on the fly, store attn once,
// and WMMA-accumulate out = attn @ h. One wave per (head, 16-row) tile.
// ---------------------------------------------------------------------------
__global__ void gat_attn_out_kernel(const int* __restrict__ adj, const float* __restrict__ h,
                                    const float* __restrict__ ssrc, const float4* __restrict__ pk,
                                    const float* __restrict__ C1, const float* __restrict__ C5,
                                    float* __restrict__ out, float* __restrict__ attn) {
  const int head = blockIdx.y;
  const int wave = threadIdx.x >> 5;
  const int lane = threadIdx.x & 31;
  const int half = lane >> 4;
  const int l    = lane & 15;
  const int n0   = blockIdx.x * 64 + wave * 16;
  const int row  = n0 + l;

  const float th = -ssrc[head * N_NODES + row];
  const float c1 = C1[head * N_NODES + row];
  const float c5 = C5[head * N_NODES + row];

  const int*    arow = adj + (size_t)row * N_NODES;
  const float4* pkh  = pk + (size_t)head * N_NODES;
  const float*  hb   = h + (size_t)head * N_NODES * OUT_F;
  float* attnrow     = attn + ((size_t)head * N_NODES + row) * N_NODES;

  v8f acc0 = {}, acc1 = {}, acc2 = {}, acc3 = {};
  for (int k = 0; k < N_NODES; k += 4) {
    const int ka = k + half * 2;
    const int2   av = *(const int2*)(arow + ka);
    const float4 p0 = pkh[ka];
    const float4 p1 = pkh[ka + 1];
    const float a0 = av.x ? ((p0.x >= th) ? c1 * p0.y : c5 * p0.z) : 0.f;
    const float a1 = av.y ? ((p1.x >= th) ? c1 * p1.y : c5 * p1.z) : 0.f;
    v2f afrag; afrag.x = a0; afrag.y = a1;

    const float* hk0 = hb + (size_t)ka * OUT_F + l;
    const float* hk1 = hk0 + OUT_F;
    v2f b0; b0.x = hk0[0];  b0.y = hk1[0];
    v2f b1; b1.x = hk0[16]; b1.y = hk1[16];
    v2f b2; b2.x = hk0[32]; b2.y = hk1[32];
    v2f b3; b3.x = hk0[48]; b3.y = hk1[48];
    acc0 = wmma_f32_16x16x4(afrag, b0, acc0);
    acc1 = wmma_f32_16x16x4(afrag, b1, acc1);
    acc2 = wmma_f32_16x16x4(afrag, b2, acc2);
    acc3 = wmma_f32_16x16x4(afrag, b3, acc3);

    float2 st; st.x = a0; st.y = a1;
    *(float2*)(attnrow + ka) = st;        // normalized attention (output #2)
  }
  for (int r = 0; r < 8; ++r) {
    const int m = n0 + r + half * 8;
    float* orow = out + (size_t)m * (HEADS * OUT_F) + head * OUT_F + l;
    orow[0]  = acc0[r];
    orow[16] = acc1[r];
    orow[32] = acc2[r];
    orow[48] = acc3[r];
  }
}

extern "C" void kernel_launch(void* const* d_in, const int* in_sizes, int n_in,
                              void* d_out, int out_size, void* d_ws, size_t ws_size,
                              hipStream_t stream) {
  const float* x     = (const float*)d_in[0];
  const int*   adj   = (const int*)d_in[1];
  const float* W     = (const float*)d_in[2];
  const float* a_src = (const float*)d_in[3];
  const float* a_dst = (const float*)d_in[4];

  float* out  = (float*)d_out;                                   // [N, HEADS*OUT_F]
  float* attn = out + (size_t)N_NODES * (HEADS * OUT_F);         // [HEADS, N, N]

  float*  ws   = (float*)d_ws;
  float*  h    = ws;                                             // HEADS*N*OUT_F floats
  float4* pk   = (float4*)(h + (size_t)HEADS * N_NODES * OUT_F); // HEADS*N float4 (16B aligned)
  float*  ssrc = (float*)(pk + (size_t)HEADS * N_NODES);         // HEADS*N
  float*  C1   = ssrc + HEADS * N_NODES;
  float*  C5   = C1 + HEADS * N_NODES;

  gat_h_kernel<<<dim3(N_NODES / 64, HEADS), 128, 0, stream>>>(x, W, h);
  gat_scores_kernel<<<(HEADS * N_NODES + 255) / 256, 256, 0, stream>>>(h, a_src, a_dst,
                                                                       ssrc, pk);
  gat_rowstats_kernel<<<N_NODES, 256, 0, stream>>>(adj, ssrc, pk, C1, C5);
  gat_attn_out_kernel<<<dim3(N_NODES / 64, HEADS), 128, 0, stream>>>(adj, h, ssrc, pk,
                                                                     C1, C5, out, attn);
}